// PyGTemporalGNN_70763881169206
// MI455X (gfx1250) — compile-verified
//
#include <hip/hip_runtime.h>

// ---------- WMMA vector types (f32 16x16x4) ----------
typedef __attribute__((ext_vector_type(2))) float v2f;
typedef __attribute__((ext_vector_type(8))) float v8f;

#define D_EMB 64
#define D_H   64
#define D_TXF 16
#define D_TX  32
#define D_Z   160   // 2*D_H + D_TX
#define BN_EPS 1e-5f
#define W_STRIDE 80 // LDS row stride for W tile: drow=2 -> 160 words == 32 mod 64 banks

// =====================================================================
// 1) init: ACC = 0 (N*64), deg = 1.0 (self loop weight)
// =====================================================================
__global__ void init_kernel(float* __restrict__ acc, float* __restrict__ deg,
                            long n64, int n) {
  long i = (long)blockIdx.x * blockDim.x + threadIdx.x;
  if (i < n64) acc[i] = 0.0f;
  if (i < n)   deg[i] = 1.0f;
}

// =====================================================================
// 2) degree accumulation over edges
// =====================================================================
__global__ void deg_accum_kernel(const int* __restrict__ dst,
                                 const float* __restrict__ ew,
                                 float* __restrict__ deg, long E) {
  long e = (long)blockIdx.x * blockDim.x + threadIdx.x;
  if (e < E) atomicAdd(&deg[dst[e]], ew[e]);
}

// =====================================================================
// 3) dinv = rsqrt(max(deg,1e-12)) (deg>0 guaranteed by self loop)
// =====================================================================
__global__ void dinv_kernel(const float* __restrict__ deg,
                            float* __restrict__ dinv, int n) {
  int i = blockIdx.x * blockDim.x + threadIdx.x;
  if (i < n) {
    float d = deg[i];
    dinv[i] = d > 0.0f ? rsqrtf(fmaxf(d, 1e-12f)) : 0.0f;
  }
}

// =====================================================================
// 4) X = H @ W  (H: n x 64, W: 64 x 64) using V_WMMA_F32_16X16X4_F32.
//    If ids != nullptr, row r of H is src[ids[r]*64 ...] (embedding gather).
//    W is staged in LDS once per block (padded stride to de-conflict the
//    two half-wave row reads). One wave = 16-row strip x 64 cols
//    (4 accumulators), K-loop 16 x K=4. Full-tile stores are unguarded.
// =====================================================================
__global__ __launch_bounds__(256) void gcn_gemm64_kernel(
    const float* __restrict__ src, const int* __restrict__ ids,
    const float* __restrict__ W, float* __restrict__ X, int n) {
  __shared__ float Wl[D_EMB * W_STRIDE];

  // cooperative load of W (64x64) into LDS, padded rows
  {
    int t = threadIdx.x;            // 256 threads, 16 floats each (4x float4)
#pragma unroll
    for (int q = 0; q < 4; ++q) {
      int idx = (t + q * 256) * 4;  // flat element in 64x64
      int row = idx >> 6;
      int col = idx & 63;
      const float4 wv = *(const float4*)(W + idx);
      float* p = &Wl[row * W_STRIDE + col];
      p[0] = wv.x; p[1] = wv.y; p[2] = wv.z; p[3] = wv.w;
    }
  }
  __syncthreads();

  int lane = threadIdx.x & 31;
  int wave = threadIdx.x >> 5;
  int rowTile = blockIdx.x * 8 + wave;
  int nTiles = (n + 15) >> 4;
  if (rowTile >= nTiles) return;   // uniform per wave; after the only barrier

  int m  = lane & 15;              // M index for A, N index for B/D
  int hi = lane >> 4;              // lane half selects K pair / row+8
  int row = rowTile * 16 + m;
  int gr  = row < n ? row : n - 1; // clamp (garbage rows never stored)
  long base = ids ? (long)ids[gr] * D_EMB : (long)gr * D_EMB;

  v8f acc[4] = {};
  for (int k0 = 0; k0 < D_EMB; k0 += 4) {
    v2f a;
    a.x = src[base + k0 + 2 * hi];
    a.y = src[base + k0 + 2 * hi + 1];
#pragma unroll
    for (int nt = 0; nt < 4; ++nt) {
      v2f b;
      b.x = Wl[(k0 + 2 * hi) * W_STRIDE + nt * 16 + m];
      b.y = Wl[(k0 + 2 * hi + 1) * W_STRIDE + nt * 16 + m];
      acc[nt] = __builtin_amdgcn_wmma_f32_16x16x4_f32(
          false, a, false, b, (short)0, acc[nt], false, false);
    }
  }

  if (rowTile * 16 + 16 <= n) {
    // fast path: whole tile in range, unguarded coalesced stores
#pragma unroll
    for (int nt = 0; nt < 4; ++nt) {
#pragma unroll
      for (int i = 0; i < 8; ++i) {
        int r = rowTile * 16 + i + 8 * hi;
        X[(long)r * D_H + nt * 16 + m] = acc[nt][i];
      }
    }
  } else {
#pragma unroll
    for (int nt = 0; nt < 4; ++nt) {
#pragma unroll
      for (int i = 0; i < 8; ++i) {
        int r = rowTile * 16 + i + 8 * hi;
        if (r < n) X[(long)r * D_H + nt * 16 + m] = acc[nt][i];
      }
    }
  }
}

// =====================================================================
// 5) edge scatter: ACC[dst] += X[src] * (dinv[s]*w*dinv[d])
//    16 threads per edge, each thread: float4 load + 4 f32 atomics.
// =====================================================================
__global__ __launch_bounds__(256) void edge_scatter_kernel(
    const int* __restrict__ src, const int* __restrict__ dst,
    const float* __restrict__ ew, const float* __restrict__ dinv,
    const float* __restrict__ X, float* __restrict__ ACC, long E) {
  long t = (long)blockIdx.x * blockDim.x + threadIdx.x;
  long e = t >> 4;
  if (e >= E) return;
  int f = (int)(t & 15) * 4;
  int s = src[e], d = dst[e];
  float norm = dinv[s] * ew[e] * dinv[d];
  const float4 xv = *(const float4*)(X + (long)s * D_H + f);
  float* ap = ACC + (long)d * D_H + f;
  atomicAdd(ap + 0, xv.x * norm);
  atomicAdd(ap + 1, xv.y * norm);
  atomicAdd(ap + 2, xv.z * norm);
  atomicAdd(ap + 3, xv.w * norm);
}

// =====================================================================
// 6) H = relu(ACC + X*dinv^2 + b); optionally re-zero ACC for next layer
// =====================================================================
__global__ void finish_layer_kernel(float* __restrict__ acc,
                                    const float* __restrict__ X,
                                    const float* __restrict__ dinv,
                                    const float* __restrict__ b,
                                    float* __restrict__ Hout,
                                    int resetAcc, int n) {
  long idx = (long)blockIdx.x * blockDim.x + threadIdx.x;
  if (idx >= (long)n * D_H) return;
  int node = (int)(idx >> 6);
  int f = (int)(idx & 63);
  float di = dinv[node];
  float v = acc[idx] + X[idx] * di * di + b[f];
  Hout[idx] = fmaxf(v, 0.0f);
  if (resetAcc) acc[idx] = 0.0f;
}

// =====================================================================
// 7) MLP head: z = [h[send], h[recv], relu(tx@Wtx+btx)] (16 x 160 in LDS)
//    -> WMMA 160x64 -> BN+ReLU -> WMMA 64x32 -> BN+ReLU -> dot 32x1.
//    One wave per 16 target rows; 4 waves / block; 40 KB LDS / block.
// =====================================================================
__global__ __launch_bounds__(128) void mlp_head_kernel(
    const float* __restrict__ H, const int* __restrict__ send,
    const int* __restrict__ recv, const int* __restrict__ ptr,
    const float* __restrict__ txf, const float* __restrict__ Wtx,
    const float* __restrict__ btx, const float* __restrict__ Wc1,
    const float* __restrict__ bc1, const float* __restrict__ g1,
    const float* __restrict__ beta1, const float* __restrict__ Wc2,
    const float* __restrict__ bc2, const float* __restrict__ g2,
    const float* __restrict__ beta2, const float* __restrict__ Wc3,
    const float* __restrict__ bc3, float* __restrict__ out, int Bn) {
  __shared__ float lds[4 * 16 * D_Z];
  int lane = threadIdx.x & 31;
  int wave = threadIdx.x >> 5;
  float* zl = lds + wave * 16 * D_Z;
  int tile = blockIdx.x * 4 + wave;
  const float bnscale = rsqrtf(1.0f + BN_EPS);

  // ---- build z tile (16 x 160) in LDS ----
  for (int idx = lane; idx < 16 * D_Z; idx += 32) {
    int rl = idx / D_Z;
    int col = idx - rl * D_Z;
    int grow = tile * 16 + rl;
    if (grow >= Bn) grow = Bn - 1;  // clamp; final store is guarded
    int off = ptr[grow];
    float val;
    if (col < D_H) {
      int sidx = send[grow] + off;
      val = H[(long)sidx * D_H + col];
    } else if (col < 2 * D_H) {
      int ridx = recv[grow] + off;
      val = H[(long)ridx * D_H + (col - D_H)];
    } else {
      int c = col - 2 * D_H;
      float s = btx[c];
#pragma unroll
      for (int j = 0; j < D_TXF; ++j)
        s += txf[(long)grow * D_TXF + j] * Wtx[j * D_TX + c];
      val = fmaxf(s, 0.0f);
    }
    zl[idx] = val;
  }
  __syncthreads();

  int m = lane & 15;
  int hi = lane >> 4;

  // ---- GEMM1: z(16x160) @ Wc1(160x64) ----
  v8f a1[4] = {};
  for (int k0 = 0; k0 < D_Z; k0 += 4) {
    v2f a;
    a.x = zl[m * D_Z + k0 + 2 * hi];
    a.y = zl[m * D_Z + k0 + 2 * hi + 1];
#pragma unroll
    for (int nt = 0; nt < 4; ++nt) {
      v2f b;
      b.x = Wc1[(k0 + 2 * hi) * D_H + nt * 16 + m];
      b.y = Wc1[(k0 + 2 * hi + 1) * D_H + nt * 16 + m];
      a1[nt] = __builtin_amdgcn_wmma_f32_16x16x4_f32(
          false, a, false, b, (short)0, a1[nt], false, false);
    }
  }
  __syncthreads();
  // BN1 + ReLU -> y1 (16 x 64) back into LDS
#pragma unroll
  for (int nt = 0; nt < 4; ++nt) {
    int col = nt * 16 + m;
    float sc = g1[col] * bnscale;
    float bb = beta1[col];
    float bi = bc1[col];
#pragma unroll
    for (int i = 0; i < 8; ++i) {
      int r = i + 8 * hi;
      float v = (a1[nt][i] + bi) * sc + bb;
      zl[r * D_H + col] = fmaxf(v, 0.0f);
    }
  }
  __syncthreads();

  // ---- GEMM2: y1(16x64) @ Wc2(64x32) ----
  v8f a2[2] = {};
  for (int k0 = 0; k0 < D_H; k0 += 4) {
    v2f a;
    a.x = zl[m * D_H + k0 + 2 * hi];
    a.y = zl[m * D_H + k0 + 2 * hi + 1];
#pragma unroll
    for (int nt = 0; nt < 2; ++nt) {
      v2f b;
      b.x = Wc2[(k0 + 2 * hi) * (D_H / 2) + nt * 16 + m];
      b.y = Wc2[(k0 + 2 * hi + 1) * (D_H / 2) + nt * 16 + m];
      a2[nt] = __builtin_amdgcn_wmma_f32_16x16x4_f32(
          false, a, false, b, (short)0, a2[nt], false, false);
    }
  }
  __syncthreads();
  // BN2 + ReLU -> y2 (16 x 32) into LDS
#pragma unroll
  for (int nt = 0; nt < 2; ++nt) {
    int col = nt * 16 + m;
    float sc = g2[col] * bnscale;
    float bb = beta2[col];
    float bi = bc2[col];
#pragma unroll
    for (int i = 0; i < 8; ++i) {
      int r = i + 8 * hi;
      float v = (a2[nt][i] + bi) * sc + bb;
      zl[r * (D_H / 2) + col] = fmaxf(v, 0.0f);
    }
  }
  __syncthreads();

  // ---- final: out = y2 @ Wc3 (32 -> 1) ----
  if (lane < 16) {
    int grow = tile * 16 + lane;
    if (grow < Bn) {
      float v = bc3[0];
#pragma unroll
      for (int k = 0; k < D_H / 2; ++k) v += zl[lane * (D_H / 2) + k] * Wc3[k];
      out[grow] = v;
    }
  }
}

// =====================================================================
// host launch
// =====================================================================
extern "C" void kernel_launch(void* const* d_in, const int* in_sizes, int n_in,
                              void* d_out, int out_size, void* d_ws,
                              size_t ws_size, hipStream_t stream) {
  const int*   ids   = (const int*)d_in[0];
  const int*   ei    = (const int*)d_in[1];
  const float* ew    = (const float*)d_in[2];
  const int*   send  = (const int*)d_in[3];
  const int*   recv  = (const int*)d_in[4];
  const int*   ptr   = (const int*)d_in[5];
  const float* txf   = (const float*)d_in[6];
  const float* emb   = (const float*)d_in[7];
  const float* W0    = (const float*)d_in[8];
  const float* b0    = (const float*)d_in[9];
  const float* W1    = (const float*)d_in[10];
  const float* b1    = (const float*)d_in[11];
  const float* Wtx   = (const float*)d_in[12];
  const float* btx   = (const float*)d_in[13];
  const float* Wc1   = (const float*)d_in[14];
  const float* bc1   = (const float*)d_in[15];
  const float* g1    = (const float*)d_in[16];
  const float* beta1 = (const float*)d_in[17];
  const float* Wc2   = (const float*)d_in[18];
  const float* bc2   = (const float*)d_in[19];
  const float* g2    = (const float*)d_in[20];
  const float* beta2 = (const float*)d_in[21];
  const float* Wc3   = (const float*)d_in[22];
  const float* bc3   = (const float*)d_in[23];

  int  n  = in_sizes[0];
  long E  = (long)in_sizes[1] / 2;
  int  Bn = in_sizes[3];
  const int* srcI = ei;
  const int* dstI = ei + E;

  long n64 = (long)n * D_H;
  float* X    = (float*)d_ws;          // n x 64
  float* ACC  = X + n64;               // n x 64
  float* Hbuf = ACC + n64;             // n x 64
  float* deg  = Hbuf + n64;            // n
  float* dinv = deg + n;               // n

  int nTiles = (n + 15) / 16;
  int gemmBlocks = (nTiles + 7) / 8;
  long scatterThreads = E * 16;

  // degrees + normalization
  init_kernel<<<(int)((n64 + 255) / 256), 256, 0, stream>>>(ACC, deg, n64, n);
  deg_accum_kernel<<<(int)((E + 255) / 256), 256, 0, stream>>>(dstI, ew, deg, E);
  dinv_kernel<<<(n + 255) / 256, 256, 0, stream>>>(deg, dinv, n);

  // GCN layer 0 (embedding gather fused into GEMM A read)
  gcn_gemm64_kernel<<<gemmBlocks, 256, 0, stream>>>(emb, ids, W0, X, n);
  edge_scatter_kernel<<<(int)((scatterThreads + 255) / 256), 256, 0, stream>>>(
      srcI, dstI, ew, dinv, X, ACC, E);
  finish_layer_kernel<<<(int)((n64 + 255) / 256), 256, 0, stream>>>(
      ACC, X, dinv, b0, Hbuf, /*resetAcc=*/1, n);

  // GCN layer 1
  gcn_gemm64_kernel<<<gemmBlocks, 256, 0, stream>>>(Hbuf, nullptr, W1, X, n);
  edge_scatter_kernel<<<(int)((scatterThreads + 255) / 256), 256, 0, stream>>>(
      srcI, dstI, ew, dinv, X, ACC, E);
  finish_layer_kernel<<<(int)((n64 + 255) / 256), 256, 0, stream>>>(
      ACC, X, dinv, b1, Hbuf, /*resetAcc=*/0, n);

  // MLP head
  int mlpBlocks = (Bn + 63) / 64;
  mlp_head_kernel<<<mlpBlocks, 128, 0, stream>>>(
      Hbuf, send, recv, ptr, txf, Wtx, btx, Wc1, bc1, g1, beta1, Wc2, bc2, g2,
      beta2, Wc3, bc3, (float*)d_out, Bn);
}